// CU_GCN_23493471109168
// MI455X (gfx1250) — compile-verified
//
#include <hip/hip_runtime.h>
#include <hip/hip_bf16.h>
#include <math.h>

typedef float v2f __attribute__((ext_vector_type(2)));
typedef float v8f __attribute__((ext_vector_type(8)));

#define NV    256
#define NB    64
#define NEDGE (NV*NV)
#define ALPHA_C 0.1f
#define COEF_C  0.45f
#define TEMP_C  0.6f

// ---------------- workspace layout (floats) ----------------
// [0      , 65536 )  W        (256x256)
// [65536  , 65792 )  dis      (256)
// [65792  , 131328)  A        (256x256)
// [131328 , 196864)  Mt0      (256x256)  = (z0 o A)^T
// [196864 , 262400)  Mt1
// [262400 , 327936)  Mt2
// [327936 , 327939)  logits   (3)

// ---- kernel 1: symmetric W from tril params + dis = deg^-1/2 ----
__global__ __launch_bounds__(NV) void build_w(const float* __restrict__ ewp,
                                              float* __restrict__ W,
                                              float* __restrict__ dis) {
    __shared__ float red[NV];
    const int i = blockIdx.x, j = threadIdx.x;
    const int hi = i > j ? i : j;
    const int lo = i > j ? j : i;
    const float w = ewp[(hi * (hi + 1)) / 2 + lo];
    W[i * NV + j] = w;
    red[j] = fabsf(w);
    __syncthreads();
    for (int s = NV / 2; s > 0; s >>= 1) {
        if (j < s) red[j] += red[j + s];
        __syncthreads();
    }
    if (j == 0) {
        const float deg = red[0];
        dis[i] = (deg > 0.f) ? (1.0f / sqrtf(deg)) : 0.f;
    }
}

// ---- kernel 2: per-layer scalars (a,b,pi,logits) + kld_loss + drop_rates ----
__device__ __forceinline__ float softplusf_(float v) {
    return (v > 20.f) ? v : log1pf(expf(v));
}
__device__ __forceinline__ float digammaf_(float x) {
    float r = 0.f;
    while (x < 6.f) { r -= 1.f / x; x += 1.f; }
    const float inv = 1.f / x, inv2 = inv * inv;
    return r + logf(x) - 0.5f * inv
           - inv2 * (1.f / 12.f - inv2 * (1.f / 120.f - inv2 * (1.f / 252.f)));
}
__global__ void scalars_kernel(const float* __restrict__ a_uc,
                               const float* __restrict__ b_uc,
                               const float* __restrict__ u_pi,
                               float* __restrict__ logits,
                               float* __restrict__ out) {
    if (threadIdx.x == 0 && blockIdx.x == 0) {
        const float euler = 0.577215664901532f;
        float kld_sum = 0.f;
        for (int l = 0; l < 3; ++l) {
            const float a  = softplusf_(fmaxf(a_uc[l], -10.f));
            const float bb = softplusf_(fminf(fmaxf(b_uc[l], -10.f), 50.f));
            const float up = fminf(fmaxf(u_pi[l], 1e-6f), 1.f - 1e-6f);
            const float pi = powf(1.f - powf(up, 1.f / bb), 1.f / a);
            logits[l]    = logf(pi) - log1pf(-pi);
            out[193 + l] = pi;                       // drop_rates
            kld_sum += (1.f - 0.8f / a) * (-euler - digammaf_(bb) - 1.f / bb)
                     + logf(a * bb + 1e-10f) - logf(0.8f) - (bb - 1.f) / bb;
        }
        out[192] = kld_sum;                          // kld_loss
    }
}

// ---- kernel 3: A = dis_i * W * dis_j ; Mt_l = (z_l o A)^T (pre-transposed) ----
__global__ __launch_bounds__(NV) void build_adj(const float* __restrict__ W,
                                                const float* __restrict__ dis,
                                                const float* __restrict__ u_rb,
                                                const float* __restrict__ logits,
                                                float* __restrict__ A,
                                                float* __restrict__ Mt0,
                                                float* __restrict__ Mt1,
                                                float* __restrict__ Mt2) {
    const int i = blockIdx.x, j = threadIdx.x;
    const int e = i * NV + j;
    const float a = dis[i] * W[e] * dis[j];
    A[e] = a;
    float* Mt[3] = {Mt0, Mt1, Mt2};
#pragma unroll
    for (int l = 0; l < 3; ++l) {
        const float ur = fminf(fmaxf(u_rb[l * NEDGE + e], 1e-6f), 1.f - 1e-6f);
        const float t  = (logits[l] + logf(ur) - log1pf(-ur)) / TEMP_C;
        const float z  = 1.f / (1.f + expf(-t));
        Mt[l][j * NV + i] = z * a;                   // transposed store
    }
}

// ---- kernel 4: fused 5-layer conv chain (f32 WMMA) + linear + pool + fc ----
__global__ __launch_bounds__(512) void gcn_main(const float* __restrict__ x,
                                                const float* __restrict__ lin_w,
                                                const float* __restrict__ lin_b,
                                                const float* __restrict__ fc_w,
                                                const float* __restrict__ fc_b,
                                                const float* __restrict__ A,
                                                const float* __restrict__ Mt0,
                                                const float* __restrict__ Mt1,
                                                const float* __restrict__ Mt2,
                                                float* __restrict__ out) {
    __shared__ float Xb[2][NV * 16];   // 256 rows x 16 padded features, ping-pong
    __shared__ float pp[4][128];       // pooling partials
    const int b   = blockIdx.x;
    const int tid = threadIdx.x;

    // load X (pad features 5 -> 16 with zeros; zeros are invariant through chain)
    for (int idx = tid; idx < NV * 16; idx += 512) {
        const int n = idx >> 4, f = idx & 15;
        Xb[0][idx] = (f < 5) ? x[(b * NV + n) * 5 + f] : 0.f;
    }
    __syncthreads();

    const float* mats[5] = {Mt0, A, Mt1, A, Mt2};   // chain: z0A^T, A, z1A^T(+relu), A, z2A^T
    const int wv   = tid >> 5;     // wave 0..15 -> output row tile
    const int lane = tid & 31;
    const int m    = lane & 15;    // A-row / B-col / D-col within tile
    const int h    = lane >> 4;    // lane half -> K pairing per ISA layout

    int cur = 0;
    for (int cv = 0; cv < 5; ++cv) {
        const float* Arow = mats[cv] + (wv * 16 + m) * NV;  // row of (pre-transposed) M
        v8f acc = {};
#pragma unroll 4
        for (int kb = 0; kb < NV; kb += 4) {
            v2f av, bv;
            // A operand: 16x4 f32 tile, lane(m,h): K = kb+2h, kb+2h+1 (contiguous pair)
            av.x = Arow[kb + 2 * h];
            av.y = Arow[kb + 2 * h + 1];
            // B operand: 4x16 f32 tile from LDS X
            bv.x = Xb[cur][(kb + 2 * h) * 16 + m];
            bv.y = Xb[cur][(kb + 2 * h + 1) * 16 + m];
            acc = __builtin_amdgcn_wmma_f32_16x16x4_f32(
                false, av, false, bv, (short)0, acc, false, false);
        }
        const int nxt = cur ^ 1;
#pragma unroll
        for (int r = 0; r < 8; ++r) {
            const int row = wv * 16 + r + 8 * h;            // D row mapping (8 VGPRs)
            float v = COEF_C * acc[r] + ALPHA_C * Xb[cur][row * 16 + m];
            if (cv == 2) v = fmaxf(v, 0.f);                 // relu on x3 layer
            Xb[nxt][row * 16 + m] = v;
        }
        cur = nxt;
        __syncthreads();
    }

    // out = relu(O @ lin_w + lin_b); pooled = column-sum over 256 nodes
    const int c = tid & 127;
    const int g = tid >> 7;                                 // 0..3
    float wcol[5];
#pragma unroll
    for (int f = 0; f < 5; ++f) wcol[f] = lin_w[f * 128 + c];
    const float bias = lin_b[c];
    float part = 0.f;
    for (int n = g; n < NV; n += 4) {
        const float* o = &Xb[cur][n * 16];
        float s = bias;
#pragma unroll
        for (int f = 0; f < 5; ++f) s += o[f] * wcol[f];
        part += fmaxf(s, 0.f);
    }
    pp[g][c] = part;
    __syncthreads();
    if (tid < 128) pp[0][tid] = pp[0][tid] + pp[1][tid] + pp[2][tid] + pp[3][tid];
    __syncthreads();
    if (tid < 3) {
        float acc = fc_b[tid];
        for (int k = 0; k < 128; ++k) acc += pp[0][k] * fc_w[k * 3 + tid];
        out[b * 3 + tid] = acc;                             // output (B x 3)
    }
}

extern "C" void kernel_launch(void* const* d_in, const int* in_sizes, int n_in,
                              void* d_out, int out_size, void* d_ws, size_t ws_size,
                              hipStream_t stream) {
    const float* x     = (const float*)d_in[0];
    const float* ewp   = (const float*)d_in[1];
    const float* a_uc  = (const float*)d_in[2];
    const float* b_uc  = (const float*)d_in[3];
    const float* u_pi  = (const float*)d_in[4];
    const float* u_rb  = (const float*)d_in[5];
    const float* lin_w = (const float*)d_in[6];
    const float* lin_b = (const float*)d_in[7];
    const float* fc_w  = (const float*)d_in[8];
    const float* fc_b  = (const float*)d_in[9];
    // d_in[10] edge_index, d_in[11] batch: dense structure known at compile time
    float* out = (float*)d_out;
    float* ws  = (float*)d_ws;

    float* W      = ws;
    float* dis    = ws + 65536;
    float* A      = ws + 65792;
    float* Mt0    = ws + 131328;
    float* Mt1    = ws + 196864;
    float* Mt2    = ws + 262400;
    float* logits = ws + 327936;

    build_w<<<NV, NV, 0, stream>>>(ewp, W, dis);
    scalars_kernel<<<1, 32, 0, stream>>>(a_uc, b_uc, u_pi, logits, out);
    build_adj<<<NV, NV, 0, stream>>>(W, dis, u_rb, logits, A, Mt0, Mt1, Mt2);
    gcn_main<<<NB, 512, 0, stream>>>(x, lin_w, lin_b, fc_w, fc_b,
                                     A, Mt0, Mt1, Mt2, out);
}